// Net_66855460929710
// MI455X (gfx1250) — compile-verified
//
#include <hip/hip_runtime.h>

typedef float v2f __attribute__((ext_vector_type(2)));
typedef float v8f __attribute__((ext_vector_type(8)));

#define NTRAIN 200000
#define NJ     768                         // B(256) * 3 output columns
#define JT     48                          // NJ / 16 column tiles
#define CHUNKS 50                          // 200000 = 50 * 4000 exactly
#define CHUNK  (NTRAIN / CHUNKS)           // 4000 = 8 waves * 125 iters * 4 n

// ---------------------------------------------------------------------------
// Kernel 1: xw_flat[j] = x[b] . W[d],  j = b*3 + d   (768 values)
// ---------------------------------------------------------------------------
__global__ void xw_kernel(const float* __restrict__ x, const float* __restrict__ W,
                          float* __restrict__ xw) {
  int j = threadIdx.x;                     // one block, 768 threads
  int b = j / 3, d = j % 3;
  float s = 0.f;
#pragma unroll
  for (int k = 0; k < 4; ++k) s += x[b * 4 + k] * W[d * 4 + k];
  xw[j] = s;
}

// ---------------------------------------------------------------------------
// Kernel 2: per (j-tile, n-chunk) block, accumulate up/down via WMMA f32 K=4.
//   A (16x4):  row0 = Y[n..n+3], row1 = 1, else 0
//   B (4x16):  K[n+k][j0+jj] = exp(-2*(proj-xw)^2)   (H = 0.5)
//   D row0 = partial up, D row1 = partial down (exact f32 accumulate)
// Chunking divides N exactly -> no guards, no divergence, EXEC always full.
// ---------------------------------------------------------------------------
__global__ __launch_bounds__(256) void nw_kernel(const float* __restrict__ tX,
                                                 const float* __restrict__ Y,
                                                 const float* __restrict__ W,
                                                 const float* __restrict__ xw,
                                                 float* __restrict__ part) {
  __shared__ float red[8][32];

  const int jt   = blockIdx.x;             // 0..47
  const int chnk = blockIdx.y;             // 0..49
  // wave id as an SGPR so loop bounds are scalar (no EXEC masking around WMMA)
  const int wave = __builtin_amdgcn_readfirstlane((int)(threadIdx.x >> 5));
  const int lane = threadIdx.x & 31;
  const int jj   = lane & 15;              // B/D: N index; A: M index
  const int hi   = lane >> 4;              // selects K-pair {0,1} vs {2,3}
  const int j    = jt * 16 + jj;
  const int d    = j % 3;

  const float c_xw = xw[j];
  const float w0 = W[d * 4 + 0], w1 = W[d * 4 + 1];
  const float w2 = W[d * 4 + 2], w3 = W[d * 4 + 3];

  // Branchless A-operand masks: av = selY * y + ones
  const float selY = (jj == 0) ? 1.0f : 0.0f;
  const float ones = (jj == 1) ? 1.0f : 0.0f;

  const int nstart = chnk * CHUNK;         // scalar
  const int nend_w = nstart + CHUNK;       // scalar

  v8f acc = {0.f, 0.f, 0.f, 0.f, 0.f, 0.f, 0.f, 0.f};

#pragma unroll 5
  for (int nb = nstart + wave * 4; nb < nend_w; nb += 32) {
    const int n0 = nb + 2 * hi;            // even; lane covers n0, n0+1

    const float4 t0  = ((const float4*)tX)[n0];        // global_load_b128
    const float4 t1  = ((const float4*)tX)[n0 + 1];    // global_load_b128
    const float2 y01 = ((const float2*)Y)[n0 >> 1];    // global_load_b64

    const float p0 = t0.x * w0 + t0.y * w1 + t0.z * w2 + t0.w * w3;
    const float p1 = t1.x * w0 + t1.y * w1 + t1.z * w2 + t1.w * w3;
    const float df0 = p0 - c_xw;
    const float df1 = p1 - c_xw;
    // H = 0.5  =>  -0.5*((p-c)/H)^2 = -2*(p-c)^2
    const float k0 = __expf(-2.0f * df0 * df0);        // v_exp_f32 (TRANS)
    const float k1 = __expf(-2.0f * df1 * df1);

    v2f bv;                                 // B: lane holds rows (2*hi, 2*hi+1), col jj
    bv.x = k0;
    bv.y = k1;

    v2f av;                                 // A: lane M = jj; row0 = Y, row1 = ones
    av.x = selY * y01.x + ones;
    av.y = selY * y01.y + ones;

    acc = __builtin_amdgcn_wmma_f32_16x16x4_f32(false, av, false, bv,
                                                (short)0, acc, false, false);
  }

  // D layout: lanes 0..15 hold row M=r in VGPR r.  acc[0]=up, acc[1]=down.
  if (lane < 16) {
    red[wave][lane]      = acc[0];
    red[wave][16 + lane] = acc[1];
  }
  __syncthreads();

  if (threadIdx.x < 32) {                   // sum the 8 waves, emit block partial
    float s = 0.f;
#pragma unroll
    for (int w = 0; w < 8; ++w) s += red[w][threadIdx.x];
    part[(chnk * JT + jt) * 32 + threadIdx.x] = s;   // [0..15]=up, [16..31]=down
  }
}

// ---------------------------------------------------------------------------
// Kernel 3: deterministic reduction over chunks, final divide.
// ---------------------------------------------------------------------------
__global__ void final_kernel(const float* __restrict__ part, float* __restrict__ out) {
  int j  = threadIdx.x;                     // one block, 768 threads
  int jt = j >> 4, jj = j & 15;
  float up = 0.f, dn = 0.f;
  for (int c = 0; c < CHUNKS; ++c) {
    const float* p = part + (c * JT + jt) * 32;
    up += p[jj];
    dn += p[16 + jj];
  }
  out[j] = up / dn;                         // layout [B,3], j = b*3 + d
}

// ---------------------------------------------------------------------------
extern "C" void kernel_launch(void* const* d_in, const int* in_sizes, int n_in,
                              void* d_out, int out_size, void* d_ws, size_t ws_size,
                              hipStream_t stream) {
  const float* x  = (const float*)d_in[0];   // [256,4]
  const float* tX = (const float*)d_in[1];   // [200000,4]
  const float* Y  = (const float*)d_in[2];   // [200000]
  const float* W  = (const float*)d_in[3];   // [3,4]

  float* ws   = (float*)d_ws;
  float* xw   = ws;                          // 768 floats
  float* part = ws + NJ;                     // CHUNKS*JT*32 floats (~300 KB)

  xw_kernel<<<1, NJ, 0, stream>>>(x, W, xw);

  dim3 grid(JT, CHUNKS);
  nw_kernel<<<grid, 256, 0, stream>>>(tX, Y, W, xw, part);

  final_kernel<<<1, NJ, 0, stream>>>(part, (float*)d_out);
}